// RoIDelta_76759655514853
// MI455X (gfx1250) — compile-verified
//
#include <hip/hip_runtime.h>
#include <hip/hip_bf16.h>
#include <stdint.h>

#define BB 64
#define NN 1500
#define MM 100
#define LL 81
#define NPOS 128
#define NNEG 128

typedef unsigned int u32;
typedef u32 u32x4 __attribute__((ext_vector_type(4)));
typedef int  i32x8 __attribute__((ext_vector_type(8)));
typedef int  i32x4 __attribute__((ext_vector_type(4)));

#if defined(__has_builtin)
#if __has_builtin(__builtin_amdgcn_tensor_load_to_lds)
#define HAVE_TDM 1
#endif
#endif
#ifndef HAVE_TDM
#define HAVE_TDM 0
#endif

#if HAVE_TDM
// 1-D contiguous global->LDS DMA of `nd` dwords via the Tensor Data Mover.
// Descriptor packing per CDNA5 ISA 8.3/8.4 (verified lowering: emits
// tensor_load_to_lds with SGPR descriptor groups + s_wait_tensorcnt).
__device__ __forceinline__ void tdm_load_1d_to_lds(u32 lds_byte, const void* gsrc, u32 nd) {
  uint64_t ga = (uint64_t)(uintptr_t)gsrc;
  u32x4 g0;
  g0[0] = 1u;                                               // count=1, user descriptor
  g0[1] = lds_byte;                                         // lds_addr (bytes)
  g0[2] = (u32)(ga & 0xFFFFFFFFull);                        // global_addr[31:0]
  g0[3] = (u32)((ga >> 32) & 0x1FFFFFFull) | (2u << 30);    // global_addr[56:32], type=2
  i32x8 g1;
  g1[0] = (int)(2u << 16);                                  // data_size = 2 (4 bytes)
  g1[1] = (int)((nd & 0xFFFFu) << 16);                      // tensor_dim0[15:0]
  g1[2] = (int)((nd >> 16) | (1u << 16));                   // tensor_dim0[31:16] | tensor_dim1=1
  g1[3] = (int)((nd & 0xFFFFu) << 16);                      // tile_dim0 = nd
  g1[4] = 1;                                                // tile_dim1 = 1, tile_dim2 = 0
  g1[5] = (int)nd;                                          // tensor_dim0_stride[31:0]
  g1[6] = (int)((nd & 0xFFFFu) << 16);                      // stride0 hi=0 | stride1[15:0]
  g1[7] = (int)(nd >> 16);                                  // stride1[47:16]
  i32x4 z = {0, 0, 0, 0};
#if __clang_major__ >= 23
  i32x8 z8 = {0, 0, 0, 0, 0, 0, 0, 0};
  __builtin_amdgcn_tensor_load_to_lds(g0, g1, z, z, z8, 0); // 6-arg toolchain variant
#else
  __builtin_amdgcn_tensor_load_to_lds(g0, g1, z, z, 0);     // ROCm 7.2 / clang-22 variant
#endif
}
#endif

// ---------------- Kernel 1: per-(b,n) best IoU / argmax + masked scores ----
__global__ void __launch_bounds__(256)
iou_kernel(const float* __restrict__ roi, const float* __restrict__ gt,
           const int* __restrict__ rpos, const int* __restrict__ rneg,
           int* __restrict__ best, int* __restrict__ psc, int* __restrict__ nsc) {
  __shared__ __align__(16) float sgt[MM * 4];
  const int b = blockIdx.y;
#if HAVE_TDM
  if (threadIdx.x < 32) {            // wave 0 issues the DMA (TDM ignores EXEC)
    tdm_load_1d_to_lds((u32)(uintptr_t)(void*)sgt, gt + (size_t)b * MM * 4, MM * 4);
    __builtin_amdgcn_s_wait_tensorcnt(0);
  }
  __syncthreads();
#else
  for (int t = threadIdx.x; t < MM; t += blockDim.x) {
    const float4 v = *(const float4*)(gt + ((size_t)b * MM + t) * 4);
    *(float4*)(sgt + t * 4) = v;
  }
  __syncthreads();
#endif
  const int n = blockIdx.x * blockDim.x + threadIdx.x;
  if (n >= NN) return;
  const size_t ro = ((size_t)b * NN + n) * 4;
  const float by1 = roi[ro + 0], bx1 = roi[ro + 1], by2 = roi[ro + 2], bx2 = roi[ro + 3];
  const float barea = (by2 - by1) * (bx2 - bx1);
  float bi = -1.0f;
  int bidx = 0;
  for (int m = 0; m < MM; ++m) {
    const float4 g = *(const float4*)(sgt + m * 4);
    const float gy1 = g.x, gx1 = g.y, gy2 = g.z, gx2 = g.w;
    const float garea = (gy2 - gy1) * (gx2 - gx1);
    const float xt = fmaxf(bx1, gx1), yt = fmaxf(by1, gy1);
    const float xb = fminf(bx2, gx2), yb = fminf(by2, gy2);
    const float inter = fmaxf(xb - xt, 0.0f) * fmaxf(yb - yt, 0.0f);
    const float iou = inter / (barea + garea - inter);
    if (iou > bi) { bi = iou; bidx = m; }   // first-max semantics like jnp.argmax
  }
  const int idx = b * NN + n;
  best[idx] = bidx;
  psc[idx] = (bi > 0.5f) ? rpos[idx] : 0;                      // rand>=1 => score>0 <=> mask
  nsc[idx] = (bi < 0.5f && bi > 0.1f) ? rneg[idx] : 0;
}

// ---------------- Kernel 2: stable top-k rank selection + row code --------
// Stable rank trick: scores < 1280 < 2^11 and N-1 < 2048, so
//   key = (score << 11) | (2047 - index)
// gives rank(i) = #{ j : key_j > key_i } with exact stable-argsort semantics
// (score descending, index ascending tie-break). score>0  <=>  key > 2047.
// code = -1 (none), 0 (neg -> one-hot label 0, zero deltas),
//        0x10000|label (pos -> one-hot label + deltas)
__global__ void __launch_bounds__(256)
rank_kernel(const int* __restrict__ psc, const int* __restrict__ nsc,
            const int* __restrict__ best, const int* __restrict__ gtl,
            int* __restrict__ code) {
  __shared__ __align__(16) int sps[NN];
  __shared__ __align__(16) int sns[NN];
  const int b = blockIdx.y;
#if HAVE_TDM
  if (threadIdx.x < 32) {
    tdm_load_1d_to_lds((u32)(uintptr_t)(void*)sps, psc + (size_t)b * NN, NN);
    tdm_load_1d_to_lds((u32)(uintptr_t)(void*)sns, nsc + (size_t)b * NN, NN);
    __builtin_amdgcn_s_wait_tensorcnt(0);
  }
  __syncthreads();
#else
  for (int t = threadIdx.x; t < NN; t += blockDim.x) {
    sps[t] = psc[(size_t)b * NN + t];
    sns[t] = nsc[(size_t)b * NN + t];
  }
  __syncthreads();
#endif
  // In-place transform to stable sort keys.
  for (int t = threadIdx.x; t < NN; t += blockDim.x) {
    sps[t] = (sps[t] << 11) | (2047 - t);
    sns[t] = (sns[t] << 11) | (2047 - t);
  }
  __syncthreads();

  const int n = blockIdx.x * blockDim.x + threadIdx.x;
  if (n >= NN) return;
  const int kp = sps[n], kn = sns[n];
  bool pos = false, neg = false;
  if (kp > 2047) {                       // pos-mask candidate
    int r = 0;
    for (int j = 0; j < NN; j += 4) {    // ds_load_b128, 4 compares per DS op
      const i32x4 v = *(const i32x4*)(sps + j);
      r += (v[0] > kp) + (v[1] > kp) + (v[2] > kp) + (v[3] > kp);
    }
    pos = (r < NPOS);
  }
  if (kn > 2047) {                       // neg-mask candidate
    int r = 0;
    for (int j = 0; j < NN; j += 4) {
      const i32x4 v = *(const i32x4*)(sns + j);
      r += (v[0] > kn) + (v[1] > kn) + (v[2] > kn) + (v[3] > kn);
    }
    neg = (r < NNEG);
  }
  const int idx = b * NN + n;
  int c = -1;
  if (pos)      c = 0x10000 | gtl[b * MM + best[idx]];
  else if (neg) c = 0;
  code[idx] = c;
}

// ---------------- Kernel 3a: zero-fill 155.5 MB output (bandwidth bound) --
__global__ void __launch_bounds__(256)
zero_kernel(float* __restrict__ out, long n) {
  const long n4 = n >> 2;
  const long stride = (long)gridDim.x * blockDim.x;
  float4* __restrict__ out4 = (float4*)out;
  const float4 z = make_float4(0.0f, 0.0f, 0.0f, 0.0f);
  for (long i = (long)blockIdx.x * blockDim.x + threadIdx.x; i < n4; i += stride)
    out4[i] = z;                                           // global_store_b128
  if (blockIdx.x == 0 && threadIdx.x == 0)                 // tail (if n % 4 != 0)
    for (long k = n4 << 2; k < n; ++k) out[k] = 0.0f;
}

// ---------------- Kernel 3b: sparse scatter of one-hot labels + deltas ----
__global__ void __launch_bounds__(256)
scatter_kernel(const float* __restrict__ roi, const float* __restrict__ gt,
               const int* __restrict__ best, const int* __restrict__ code,
               float* __restrict__ deltas, float* __restrict__ labels) {
  const int i = blockIdx.x * blockDim.x + threadIdx.x;
  if (i >= BB * NN) return;
  const int c = code[i];
  if (c < 0) return;
  const int l = c & 0xFFFF;
  labels[(size_t)i * LL + l] = 1.0f;
  if (c & 0x10000) {
    const int b = i / NN;
    const float4 rb = *(const float4*)(roi + (size_t)i * 4);
    const float by1 = rb.x, bx1 = rb.y, by2 = rb.z, bx2 = rb.w;
    const float4 gb = *(const float4*)(gt + ((size_t)b * MM + best[i]) * 4);
    const float gy1 = gb.x, gx1 = gb.y, gy2 = gb.z, gx2 = gb.w;
    float bw = bx2 - bx1, bh = by2 - by1;
    const float bcx = bx1 + 0.5f * bw, bcy = by1 + 0.5f * bh;   // centers use original bw/bh
    const float gw = gx2 - gx1, gh = gy2 - gy1;
    const float gcx = gx1 + 0.5f * gw, gcy = gy1 + 0.5f * gh;
    if (bw == 0.0f) bw = 0.001f;
    if (bh == 0.0f) bh = 0.001f;
    const float dx = (gw == 0.0f) ? 0.0f : (gcx - bcx) / bw;
    const float dy = (gh == 0.0f) ? 0.0f : (gcy - bcy) / bh;
    const float dw = (gw == 0.0f) ? 0.0f : logf(gw / bw);
    const float dh = (gh == 0.0f) ? 0.0f : logf(gh / bh);
    float* d = deltas + ((size_t)i * LL + l) * 4;
    d[0] = dy / 0.1f;
    d[1] = dx / 0.1f;
    d[2] = dh / 0.2f;
    d[3] = dw / 0.2f;
  }
}

extern "C" void kernel_launch(void* const* d_in, const int* in_sizes, int n_in,
                              void* d_out, int out_size, void* d_ws, size_t ws_size,
                              hipStream_t stream) {
  const float* roi  = (const float*)d_in[0];   // [B,N,4] f32
  const float* gt   = (const float*)d_in[1];   // [B,M,4] f32
  const int*   gtl  = (const int*)d_in[2];     // [B,M] i32
  const int*   rpos = (const int*)d_in[3];     // [B,N] i32
  const int*   rneg = (const int*)d_in[4];     // [B,N] i32

  int* best = (int*)d_ws;                      // [B*N]
  int* psc  = best + BB * NN;                  // [B*N]
  int* nsc  = psc + BB * NN;                   // [B*N]
  int* code = nsc + BB * NN;                   // [B*N]

  float* deltas = (float*)d_out;                       // [B,N,81,4]
  float* labels = deltas + (size_t)BB * NN * LL * 4;   // [B,N,81]

  const dim3 grid_bn((NN + 255) / 256, BB);            // 6 x 64 = 384 blocks
  iou_kernel<<<grid_bn, 256, 0, stream>>>(roi, gt, rpos, rneg, best, psc, nsc);
  rank_kernel<<<grid_bn, 256, 0, stream>>>(psc, nsc, best, gtl, code);

  zero_kernel<<<9600, 256, 0, stream>>>((float*)d_out, (long)out_size);
  scatter_kernel<<<(BB * NN + 255) / 256, 256, 0, stream>>>(roi, gt, best, code, deltas, labels);
}